// BaseSMPLRenderer_23390391894677
// MI455X (gfx1250) — compile-verified
//
#include <hip/hip_runtime.h>
#include <hip/hip_bf16.h>
#include <math.h>

// ---------------- problem constants ----------------
static constexpr int NV     = 6890;
static constexpr int NF     = 13776;
static constexpr int TT     = 3;          // texture cube dim
static constexpr int SS     = 256;        // image side
static constexpr int PIX    = SS * SS;    // 65536
static constexpr int NTILE  = (NF + 15) / 16;      // 861 (exact)
static constexpr int NTPAD  = 864;                 // multiple of TPR
static constexpr int TPR    = 8;                   // face tiles staged per LDS round
static constexpr float EZOFF = 2.7320508075688772f; // 1/tan(30deg) + 1  (camera R == I)
static constexpr float IZ_NEAR_LIM = 10.0f;        // 1/NEAR
static constexpr float IZ_FAR_LIM  = 0.04f;        // 1/FAR

typedef __attribute__((ext_vector_type(2))) float v2f;
typedef __attribute__((ext_vector_type(8))) float v8f;

// ------------------------------------------------------------------
// Kernel 1a: per-face affine coefficients for {e0,e1,e2, inv_z}
// faceCoef[f][16] = A0,B0,C0, A1,B1,C1, A2,B2,C2, P,Q,R, den, zs0,zs1,zs2
// ------------------------------------------------------------------
__global__ void face_setup(const float* __restrict__ cam,
                           const float* __restrict__ verts,
                           const int*   __restrict__ faces,
                           float*       __restrict__ faceCoef)
{
    int f = blockIdx.x * blockDim.x + threadIdx.x;
    if (f >= NF) return;
    const float c0 = cam[0], c1 = cam[1], c2 = cam[2];
    float X[3], Y[3], Z[3];
    #pragma unroll
    for (int j = 0; j < 3; ++j) {
        int vi = faces[f * 3 + j];
        const float* v = verts + vi * 3;
        float sx = c0 * (v[0] + c1);
        float sy = c0 * (v[1] + c2);
        X[j] = sx;            // camera-space x (R == identity)
        Y[j] = -sy;           // proj flips y
        Z[j] = v[2] + EZOFF;  // z - eye_z
    }
    float det = (X[1] - X[0]) * (Y[2] - Y[0]) - (Y[1] - Y[0]) * (X[2] - X[0]);
    // e_i(px,py) = A_i + B_i*px + C_i*py
    float A0 = X[1] * Y[2] - Y[1] * X[2], B0 = Y[1] - Y[2], C0 = X[2] - X[1];
    float A1 = X[2] * Y[0] - Y[2] * X[0], B1 = Y[2] - Y[0], C1 = X[0] - X[2];
    float A2 = X[0] * Y[1] - Y[0] * X[1], B2 = Y[0] - Y[1], C2 = X[1] - X[0];
    bool  ok  = det > 1e-10f;
    float den = (fabsf(det) > 1e-10f) ? det : 1.0f;
    float zs0 = (fabsf(Z[0]) > 1e-8f) ? Z[0] : 1.0f;
    float zs1 = (fabsf(Z[1]) > 1e-8f) ? Z[1] : 1.0f;
    float zs2 = (fabsf(Z[2]) > 1e-8f) ? Z[2] : 1.0f;
    float r0 = 1.0f / zs0, r1 = 1.0f / zs1, r2 = 1.0f / zs2, rd = 1.0f / den;
    // inv_z(px,py) = P + Q*px + R*py
    float P = (A0 * r0 + A1 * r1 + A2 * r2) * rd;
    float Q = (B0 * r0 + B1 * r1 + B2 * r2) * rd;
    float R = (C0 * r0 + C1 * r1 + C2 * r2) * rd;
    if (!ok) {  // back-facing / degenerate: force e0 = -1 so it never passes
        A0 = -1.0f; B0 = 0.0f; C0 = 0.0f;
        A1 = B1 = C1 = A2 = B2 = C2 = 0.0f;
        P = Q = R = 0.0f;
    }
    float* o = faceCoef + f * 16;
    o[0] = A0; o[1] = B0; o[2]  = C0;
    o[3] = A1; o[4] = B1; o[5]  = C1;
    o[6] = A2; o[7] = B2; o[8]  = C2;
    o[9] = P;  o[10] = Q; o[11] = R;
    o[12] = den; o[13] = zs0; o[14] = zs1; o[15] = zs2;
}

// ------------------------------------------------------------------
// Kernel 1b: swizzle coefficients into WMMA 16x4 f32 A-operand layout.
// tileA[t][m][lane][2]: lane<16 -> (K0,K1)=(A,B) for face M=lane;
//                       lane>=16 -> (K2,K3)=(C,0) for face M=lane-16.
// m = 0..2 edges, m = 3 inv_z.
// ------------------------------------------------------------------
__global__ void tile_swizzle(const float* __restrict__ faceCoef,
                             float*       __restrict__ tileA)
{
    int id = blockIdx.x * blockDim.x + threadIdx.x;
    if (id >= NTPAD * 32) return;
    int t = id >> 5, l = id & 31;
    int M = l & 15;
    int face = t * 16 + M;
    bool lowK = (l < 16);
    #pragma unroll
    for (int m = 0; m < 4; ++m) {
        float A, B, C;
        if (face < NF) {
            const float* b = faceCoef + face * 16 + ((m < 3) ? m * 3 : 9);
            A = b[0]; B = b[1]; C = b[2];
        } else {  // padding faces: always-fail pattern
            A = (m == 0) ? -1.0f : 0.0f; B = 0.0f; C = 0.0f;
        }
        float* o = tileA + (((t * 4 + m) * 32) + l) * 2;
        o[0] = lowK ? A : C;
        o[1] = lowK ? B : 0.0f;
    }
}

// ------------------------------------------------------------------
// Kernel 3: per-face 3x3 bilinear grid-sample -> `textures` output
// outTex[f][ti][tj][k][c], k replicated (light == 1 since DIR_INT==0)
// ------------------------------------------------------------------
__global__ void tex_sample(const float* __restrict__ cam,
                           const float* __restrict__ verts,
                           const int*   __restrict__ faces,
                           const float* __restrict__ img,   // [3][256][256]
                           float*       __restrict__ outTex)
{
    int id = blockIdx.x * blockDim.x + threadIdx.x;
    if (id >= NF * 9) return;
    int f = id / 9, t = id % 9, ti = t / 3, tj = t % 3;
    const float ab[3] = {0.0f, 0.5f, 1.0f};
    const float c0 = cam[0], c1 = cam[1], c2 = cam[2];
    float sx[3], sy[3];
    #pragma unroll
    for (int j = 0; j < 3; ++j) {
        int vi = faces[f * 3 + j];
        const float* v = verts + vi * 3;
        sx[j] = c0 * (v[0] + c1);
        sy[j] = c0 * (v[1] + c2);
    }
    float a = ab[ti], b = ab[tj];
    float cx = sx[2] + a * (sx[0] - sx[2]) + b * (sx[1] - sx[2]);
    float cy = sy[2] + a * (sy[0] - sy[2]) + b * (sy[1] - sy[2]);
    cx = fminf(fmaxf(cx, -1.0f), 1.0f);
    cy = fminf(fmaxf(cy, -1.0f), 1.0f);
    float gx = (cx + 1.0f) * 128.0f - 0.5f;
    float gy = (cy + 1.0f) * 128.0f - 0.5f;
    float x0f = floorf(gx), y0f = floorf(gy);
    float fx = gx - x0f, fy = gy - y0f;
    int x0 = (int)x0f, y0 = (int)y0f;
    float col[3] = {0.0f, 0.0f, 0.0f};
    #pragma unroll
    for (int dy = 0; dy < 2; ++dy) {
        #pragma unroll
        for (int dx = 0; dx < 2; ++dx) {
            int xi = x0 + dx, yi = y0 + dy;
            bool valid = (xi >= 0) && (xi < 256) && (yi >= 0) && (yi < 256);
            int xc = min(max(xi, 0), 255), yc = min(max(yi, 0), 255);
            float w = (dx ? fx : 1.0f - fx) * (dy ? fy : 1.0f - fy);
            w = valid ? w : 0.0f;
            #pragma unroll
            for (int c = 0; c < 3; ++c)
                col[c] += w * img[c * PIX + yc * 256 + xc];
        }
    }
    float* o = outTex + f * 81 + ti * 27 + tj * 9;
    #pragma unroll
    for (int k = 0; k < 3; ++k)
        #pragma unroll
        for (int c = 0; c < 3; ++c)
            o[k * 3 + c] = col[c];
}

// ------------------------------------------------------------------
// Kernel 2: rasterizer. One wave = 16 pixels; loop over face tiles.
// 4x V_WMMA_F32_16X16X4_F32 per 16-face x 16-pixel tile.
// ------------------------------------------------------------------
__global__ __launch_bounds__(256) void raster(
    const float* __restrict__ tileA,    // [NTPAD][4][32][2]
    const float* __restrict__ faceCoef, // [NF][16]
    const float* __restrict__ texTab,   // [NF][3][3][3][3]
    float*       __restrict__ outImg)   // [3][256][256]
{
    __shared__ __align__(16) float lds[TPR * 256];
    const int tid  = threadIdx.x;
    const int lane = tid & 31;
    const int wave = tid >> 5;
    const int pix  = (blockIdx.x * 8 + wave) * 16 + (lane & 15);
    const int xI = pix & (SS - 1);
    const int yI = pix >> 8;
    const float fpx = (2.0f * (float)xI + 1.0f - (float)SS) * (1.0f / (float)SS);
    const float fpy = (2.0f * (float)yI + 1.0f - (float)SS) * (1.0f / (float)SS);

    // B operand (4x16, K x N): col n gets K0=1, K1=px from lane n; K2=py, K3=0 from lane n+16
    v2f bvec;
    if (lane < 16) { bvec.x = 1.0f; bvec.y = fpx; }
    else           { bvec.x = fpy;  bvec.y = 0.0f; }

    // Track max inv_z (nearest wins). Seeding with 1/FAR makes the
    // "izEff > bestIz" test subsume the far-plane check (valid iz > 0.04).
    float bestIz  = IZ_FAR_LIM;
    int   bestFid = -1;

    const int rounds = NTPAD / TPR;  // 108
    for (int rd = 0; rd < rounds; ++rd) {
        // cooperatively stage TPR tiles (8KB) into LDS
        const float4* s4 = (const float4*)(tileA + rd * (TPR * 256));
        float4* l4 = (float4*)lds;
        l4[tid]       = s4[tid];
        l4[tid + 256] = s4[tid + 256];
        if (rd + 1 < rounds)
            __builtin_prefetch(tileA + (rd + 1) * (TPR * 256), 0, 1);
        __syncthreads();
        #pragma unroll
        for (int tt = 0; tt < TPR; ++tt) {
            const int tileIdx = rd * TPR + tt;
            const v2f* tp = (const v2f*)(lds + tt * 256);
            v2f a0 = tp[0 * 32 + lane];
            v2f a1 = tp[1 * 32 + lane];
            v2f a2 = tp[2 * 32 + lane];
            v2f a3 = tp[3 * 32 + lane];
            v8f z = {};
            v8f d0 = __builtin_amdgcn_wmma_f32_16x16x4_f32(false, a0, false, bvec, (short)0, z, false, false);
            v8f d1 = __builtin_amdgcn_wmma_f32_16x16x4_f32(false, a1, false, bvec, (short)0, z, false, false);
            v8f d2 = __builtin_amdgcn_wmma_f32_16x16x4_f32(false, a2, false, bvec, (short)0, z, false, false);
            v8f d3 = __builtin_amdgcn_wmma_f32_16x16x4_f32(false, a3, false, bvec, (short)0, z, false, false);
            const int fbase = tileIdx * 16 + ((lane < 16) ? 0 : 8);
            #pragma unroll
            for (int r = 0; r < 8; ++r) {
                float e0 = d0[r], e1 = d1[r], e2 = d2[r], iz = d3[r];
                // min3 reduction for the three edge signs (one compare)
                float mAll = fminf(fminf(e0, e1), e2);
                float izEff = (fabsf(iz) > 1e-8f) ? iz : 1.0f;
                // valid & nearer: edge signs, near-plane, z-compare
                if (mAll >= 0.0f && izEff < IZ_NEAR_LIM && izEff > bestIz) {
                    bestIz = izEff;
                    bestFid = fbase + r;
                }
            }
        }
        __syncthreads();
    }

    // merge the two face-halves held by lanes n and n+16 (same pixel)
    float oIz  = __shfl_xor(bestIz, 16, 32);
    int   oFid = __shfl_xor(bestFid, 16, 32);
    if (oIz > bestIz || (oIz == bestIz && oFid >= 0 && oFid < bestFid)) {
        bestIz = oIz; bestFid = oFid;
    }

    if (lane < 16) {
        float col[3] = {0.0f, 0.0f, 0.0f};   // BG
        if (bestFid >= 0) {
            const float* fc = faceCoef + bestFid * 16;
            float e0 = fc[0] + fc[1] * fpx + fc[2] * fpy;
            float e1 = fc[3] + fc[4] * fpx + fc[5] * fpy;
            float e2 = fc[6] + fc[7] * fpx + fc[8] * fpy;
            float rden = 1.0f / fc[12];
            float bz = 1.0f / bestIz;        // perspective depth
            float w0 = e0 * rden / fc[13] * bz;
            float w1 = e1 * rden / fc[14] * bz;
            float w2 = e2 * rden / fc[15] * bz;
            float pos0 = fminf(fmaxf(w0, 0.0f), 1.0f) * (float)(TT - 1);
            float pos1 = fminf(fmaxf(w1, 0.0f), 1.0f) * (float)(TT - 1);
            float pos2 = fminf(fmaxf(w2, 0.0f), 1.0f) * (float)(TT - 1);
            int i0 = min(max((int)floorf(pos0), 0), TT - 2);
            int i1 = min(max((int)floorf(pos1), 0), TT - 2);
            int i2 = min(max((int)floorf(pos2), 0), TT - 2);
            float f0 = fminf(fmaxf(pos0 - (float)i0, 0.0f), 1.0f);
            float f1 = fminf(fmaxf(pos1 - (float)i1, 0.0f), 1.0f);
            float f2 = fminf(fmaxf(pos2 - (float)i2, 0.0f), 1.0f);
            const float* tb = texTab + bestFid * 81;
            #pragma unroll
            for (int d0i = 0; d0i < 2; ++d0i)
                #pragma unroll
                for (int d1i = 0; d1i < 2; ++d1i)
                    #pragma unroll
                    for (int d2i = 0; d2i < 2; ++d2i) {
                        float w = (d0i ? f0 : 1.0f - f0) *
                                  (d1i ? f1 : 1.0f - f1) *
                                  (d2i ? f2 : 1.0f - f2);
                        const float* c = tb + (i0 + d0i) * 27 + (i1 + d1i) * 9 + (i2 + d2i) * 3;
                        col[0] += w * c[0];
                        col[1] += w * c[1];
                        col[2] += w * c[2];
                    }
        }
        outImg[0 * PIX + pix] = col[0];
        outImg[1 * PIX + pix] = col[1];
        outImg[2 * PIX + pix] = col[2];
    }
}

// ------------------------------------------------------------------
extern "C" void kernel_launch(void* const* d_in, const int* in_sizes, int n_in,
                              void* d_out, int out_size, void* d_ws, size_t ws_size,
                              hipStream_t stream)
{
    const float* cam   = (const float*)d_in[0];
    const float* verts = (const float*)d_in[1];
    const float* uv    = (const float*)d_in[2];
    const int*   faces = (const int*)d_in[3];

    float* out    = (float*)d_out;
    float* outImg = out;                 // (1,3,256,256)
    float* outTex = out + 3 * PIX;       // (1,NF,3,3,3,3)

    float* faceCoef = (float*)d_ws;              // NF*16 floats
    float* tileA    = faceCoef + NF * 16;        // NTPAD*256 floats

    face_setup<<<(NF + 255) / 256, 256, 0, stream>>>(cam, verts, faces, faceCoef);
    tile_swizzle<<<(NTPAD * 32 + 255) / 256, 256, 0, stream>>>(faceCoef, tileA);
    tex_sample<<<(NF * 9 + 255) / 256, 256, 0, stream>>>(cam, verts, faces, uv, outTex);
    raster<<<PIX / 16 / 8, 256, 0, stream>>>(tileA, faceCoef, outTex, outImg);
}